// GAT_14276471292531
// MI455X (gfx1250) — compile-verified
//
#include <hip/hip_runtime.h>
#include <cmath>

// ---------------------------------------------------------------------------
// Types for WMMA f32 16x16x4 (CDNA5): A = 16x4 f32 (2 VGPRs), B = 4x16 f32
// (2 VGPRs), C/D = 16x16 f32 (8 VGPRs).
// ---------------------------------------------------------------------------
typedef float v2f __attribute__((ext_vector_type(2)));
typedef float v8f __attribute__((ext_vector_type(8)));

#define TPB 256
#define CDIV(a, b) (((a) + (b) - 1) / (b))

// Problem constants (from the reference)
#define N0 131072
#define N1 32768
#define N2 8192
#define N3 2048
#define HN 4
#define C_HID 64
#define C_OUT 47
#define HC 256  // HN * C_HID

// ---------------------------------------------------------------------------
// GEMM: C[M,N] = A[M,K] @ B[K,N] (+ bias[N] if bias != nullptr)
// One wave per 16x16 output tile, K stepped by 4 via v_wmma_f32_16x16x4_f32.
// Requirements (all true for this model): M % 16 == 0, K % 4 == 0.
// N may be ragged: lanes with nb >= N use a clamped address; their garbage
// stays confined to their own accumulator lanes, which are never stored, so
// the inner loop is completely branchless (EXEC all-ones throughout).
// ---------------------------------------------------------------------------
__global__ void wmma_gemm_f32(const float* __restrict__ A,
                              const float* __restrict__ B,
                              const float* __restrict__ bias,
                              float* __restrict__ C,
                              int M, int N, int K)
{
    const int wavesPerBlock = blockDim.x >> 5;
    const int wave  = blockIdx.x * wavesPerBlock + (threadIdx.x >> 5);
    const int tilesN = (N + 15) >> 4;
    const int tilesM = M >> 4;
    if (wave >= tilesM * tilesN) return;   // wave-uniform: EXEC stays all-ones

    const int tm   = wave / tilesN;
    const int tn   = wave - tm * tilesN;
    const int lane = threadIdx.x & 31;
    const int half = lane >> 4;   // 0: lanes 0-15, 1: lanes 16-31
    const int l16  = lane & 15;
    const int m0   = tm << 4;
    const int n0   = tn << 4;
    const int nb   = n0 + l16;
    const int nbc  = (nb < N) ? nb : (N - 1);   // clamped address, no masking

    v8f acc = {0.f, 0.f, 0.f, 0.f, 0.f, 0.f, 0.f, 0.f};

    const float* arow = A + (size_t)(m0 + l16) * K + (half << 1);
    const float* bcol = B + nbc + (size_t)(half << 1) * N;

    #pragma unroll 4
    for (int k0 = 0; k0 < K; k0 += 4) {
        v2f a, b;
        a.x = arow[0];           // merges into global_load_b64
        a.y = arow[1];
        b.x = bcol[0];
        b.y = bcol[(size_t)N];
        acc = __builtin_amdgcn_wmma_f32_16x16x4_f32(
            /*neg_a=*/false, a, /*neg_b=*/false, b,
            /*c_mod=*/(short)0, acc, /*reuse_a=*/false, /*reuse_b=*/false);
        arow += 4;
        bcol += (size_t)4 * N;
    }

    if (nb < N) {
        const float bv = bias ? bias[nb] : 0.f;
        #pragma unroll
        for (int v = 0; v < 8; ++v) {
            const int row = m0 + v + (half << 3);  // C/D layout: VGPR v -> M=v / v+8
            C[(size_t)row * N + nb] = acc[v] + bv;
        }
    }
}

// ---------------------------------------------------------------------------
// a[n,h] = sum_c xl[n,h,c] * att[h,c]
// ---------------------------------------------------------------------------
__global__ void attn_coef_kernel(const float* __restrict__ xl,
                                 const float* __restrict__ att,
                                 float* __restrict__ out,
                                 int n, int H, int C)
{
    const int idx = blockIdx.x * blockDim.x + threadIdx.x;
    if (idx >= n * H) return;
    const int node = idx / H;
    const int h    = idx - node * H;
    const float* row = xl + ((size_t)node * H + h) * C;
    const float* ah  = att + (size_t)h * C;
    float s = 0.f;
    for (int c = 0; c < C; ++c) s += row[c] * ah[c];
    out[idx] = s;
}

// Order-preserving float <-> uint encoding for atomicMax-based segment max.
__device__ __forceinline__ unsigned f2ord(float f) {
    unsigned u = __float_as_uint(f);
    return (u & 0x80000000u) ? ~u : (u | 0x80000000u);
}
__device__ __forceinline__ float ord2f(unsigned u) {
    unsigned b = (u & 0x80000000u) ? (u & 0x7FFFFFFFu) : ~u;
    return __uint_as_float(b);
}

// ---------------------------------------------------------------------------
// Segment max of leaky_relu(a_s[src] + a_d[dst]) over destination nodes.
// ---------------------------------------------------------------------------
__global__ void edge_max_kernel(const float* __restrict__ a_s,
                                const float* __restrict__ a_d,
                                const int* __restrict__ src,
                                const int* __restrict__ dst,
                                unsigned* __restrict__ m,
                                int E, int H)
{
    const int idx = blockIdx.x * blockDim.x + threadIdx.x;
    if (idx >= E * H) return;
    const int e = idx / H;
    const int h = idx - e * H;
    float v = a_s[(size_t)src[e] * H + h] + a_d[(size_t)dst[e] * H + h];
    v = (v >= 0.f) ? v : 0.2f * v;
    atomicMax(&m[(size_t)dst[e] * H + h], f2ord(v));
}

// ---------------------------------------------------------------------------
// ex[e,h] = exp(e_val - m[dst]); denom[dst,h] += ex
// ---------------------------------------------------------------------------
__global__ void edge_exp_kernel(const float* __restrict__ a_s,
                                const float* __restrict__ a_d,
                                const int* __restrict__ src,
                                const int* __restrict__ dst,
                                const unsigned* __restrict__ m,
                                float* __restrict__ den,
                                float* __restrict__ ex,
                                int E, int H)
{
    const int idx = blockIdx.x * blockDim.x + threadIdx.x;
    if (idx >= E * H) return;
    const int e = idx / H;
    const int h = idx - e * H;
    const int d = dst[e];
    float v = a_s[(size_t)src[e] * H + h] + a_d[(size_t)d * H + h];
    v = (v >= 0.f) ? v : 0.2f * v;
    const float exv = expf(v - ord2f(m[(size_t)d * H + h]));
    ex[idx] = exv;
    atomicAdd(&den[(size_t)d * H + h], exv);
}

// ---------------------------------------------------------------------------
// agg[dst,h,c] += (ex[e,h] / den[dst,h]) * xl[src,h,c]
// Atomic scatter; accumulator fits in the 192 MB L2.
// ---------------------------------------------------------------------------
__global__ void edge_agg_kernel(const float* __restrict__ xl,
                                const float* __restrict__ ex,
                                const float* __restrict__ den,
                                const int* __restrict__ src,
                                const int* __restrict__ dst,
                                float* __restrict__ agg,
                                long long total, int H, int C)
{
    const long long idx = (long long)blockIdx.x * blockDim.x + threadIdx.x;
    if (idx >= total) return;
    const int  c = (int)(idx % C);
    long long  t = idx / C;
    const int  h = (int)(t % H);
    const int  e = (int)(t / H);
    const int  d = dst[e];
    const float w = ex[(size_t)e * H + h] / den[(size_t)d * H + h];
    atomicAdd(&agg[((size_t)d * H + h) * C + c],
              w * xl[((size_t)src[e] * H + h) * C + c]);
}

// ---------------------------------------------------------------------------
// out = elu(agg + gat_bias[col] + skip)   (layers 1 & 2; skip already has bs)
// ---------------------------------------------------------------------------
__global__ void finalize_elu_kernel(const float* __restrict__ agg,
                                    const float* __restrict__ bias,
                                    const float* __restrict__ skip,
                                    float* __restrict__ out,
                                    long long total, int ncols)
{
    const long long idx = (long long)blockIdx.x * blockDim.x + threadIdx.x;
    if (idx >= total) return;
    const int col = (int)(idx % ncols);
    const float v = agg[idx] + bias[col] + skip[idx];
    out[idx] = (v > 0.f) ? v : expm1f(v);
}

// ---------------------------------------------------------------------------
// Layer 3: head-mean + b3 + skip, then row log_softmax over C_OUT.
// ---------------------------------------------------------------------------
__global__ void finalize3_kernel(const float* __restrict__ agg,
                                 const float* __restrict__ b3,
                                 const float* __restrict__ skip,
                                 float* __restrict__ out,
                                 int n, int H, int C)
{
    const int node = blockIdx.x * blockDim.x + threadIdx.x;
    if (node >= n) return;
    float tmp[64];
    float mx = -INFINITY;
    for (int c = 0; c < C; ++c) {
        float s = 0.f;
        for (int h = 0; h < H; ++h) s += agg[((size_t)node * H + h) * C + c];
        s = s / (float)H + b3[c] + skip[(size_t)node * C + c];
        tmp[c] = s;
        if (s > mx) mx = s;
    }
    float sum = 0.f;
    for (int c = 0; c < C; ++c) sum += expf(tmp[c] - mx);
    const float lse = mx + logf(sum);
    for (int c = 0; c < C; ++c) out[(size_t)node * C + c] = tmp[c] - lse;
}

// ---------------------------------------------------------------------------
// Host-side helper: one GAT attention + aggregation pass.
// ---------------------------------------------------------------------------
static void gat_attention(const float* xl, int n_src, int n_dst,
                          const int* src, const int* dst, int E,
                          const float* att_s, const float* att_d,
                          float* a_s, float* a_d, unsigned* m, float* den,
                          float* ex, float* agg, int C, hipStream_t stream)
{
    hipMemsetAsync(m,   0, (size_t)n_dst * HN * sizeof(unsigned), stream);
    hipMemsetAsync(den, 0, (size_t)n_dst * HN * sizeof(float),    stream);
    hipMemsetAsync(agg, 0, (size_t)n_dst * HN * C * sizeof(float), stream);

    attn_coef_kernel<<<CDIV(n_src * HN, TPB), TPB, 0, stream>>>(xl, att_s, a_s, n_src, HN, C);
    attn_coef_kernel<<<CDIV(n_dst * HN, TPB), TPB, 0, stream>>>(xl, att_d, a_d, n_dst, HN, C);
    edge_max_kernel<<<CDIV(E * HN, TPB), TPB, 0, stream>>>(a_s, a_d, src, dst, m, E, HN);
    edge_exp_kernel<<<CDIV(E * HN, TPB), TPB, 0, stream>>>(a_s, a_d, src, dst, m, den, ex, E, HN);
    const long long tot = (long long)E * HN * C;
    edge_agg_kernel<<<(unsigned)CDIV(tot, (long long)TPB), TPB, 0, stream>>>(
        xl, ex, den, src, dst, agg, tot, HN, C);
}

static void launch_gemm(const float* A, const float* B, const float* bias, float* C,
                        int M, int N, int K, hipStream_t stream)
{
    const int tiles = (M / 16) * CDIV(N, 16);
    wmma_gemm_f32<<<CDIV(tiles, TPB / 32), TPB, 0, stream>>>(A, B, bias, C, M, N, K);
}

extern "C" void kernel_launch(void* const* d_in, const int* in_sizes, int n_in,
                              void* d_out, int out_size, void* d_ws, size_t ws_size,
                              hipStream_t stream)
{
    (void)n_in; (void)out_size; (void)ws_size;

    const float* x    = (const float*)d_in[0];
    const int*   src1 = (const int*)d_in[1];
    const int*   dst1 = (const int*)d_in[2];
    const int*   src2 = (const int*)d_in[3];
    const int*   dst2 = (const int*)d_in[4];
    const int*   src3 = (const int*)d_in[5];
    const int*   dst3 = (const int*)d_in[6];
    const float* W1  = (const float*)d_in[7];
    const float* as1 = (const float*)d_in[8];
    const float* ad1 = (const float*)d_in[9];
    const float* b1  = (const float*)d_in[10];
    const float* Ws1 = (const float*)d_in[11];
    const float* bs1 = (const float*)d_in[12];
    const float* W2  = (const float*)d_in[13];
    const float* as2 = (const float*)d_in[14];
    const float* ad2 = (const float*)d_in[15];
    const float* b2  = (const float*)d_in[16];
    const float* Ws2 = (const float*)d_in[17];
    const float* bs2 = (const float*)d_in[18];
    const float* W3  = (const float*)d_in[19];
    const float* as3 = (const float*)d_in[20];
    const float* ad3 = (const float*)d_in[21];
    const float* b3  = (const float*)d_in[22];
    const float* Ws3 = (const float*)d_in[23];
    const float* bs3 = (const float*)d_in[24];

    const int E1 = in_sizes[1];
    const int E2 = in_sizes[3];
    const int E3 = in_sizes[5];
    const int F_IN = 100;
    const int NOUT3 = HN * C_OUT;  // 188

    // --- workspace bump allocator -----------------------------------------
    char* wp = (char*)d_ws;
    auto alloc = [&](size_t bytes) -> void* {
        void* r = (void*)wp;
        wp += (bytes + 255) & ~(size_t)255;
        return r;
    };
    float*    xl1   = (float*)alloc((size_t)N0 * HC * 4);
    float*    skip1 = (float*)alloc((size_t)N1 * HC * 4);
    float*    a_s1  = (float*)alloc((size_t)N0 * HN * 4);
    float*    a_d1  = (float*)alloc((size_t)N1 * HN * 4);
    unsigned* m1    = (unsigned*)alloc((size_t)N1 * HN * 4);
    float*    den1  = (float*)alloc((size_t)N1 * HN * 4);
    float*    ex1   = (float*)alloc((size_t)E1 * HN * 4);
    float*    agg1  = (float*)alloc((size_t)N1 * HC * 4);
    float*    h1    = (float*)alloc((size_t)N1 * HC * 4);

    float*    xl2   = (float*)alloc((size_t)N1 * HC * 4);
    float*    skip2 = (float*)alloc((size_t)N2 * HC * 4);
    float*    a_s2  = (float*)alloc((size_t)N1 * HN * 4);
    float*    a_d2  = (float*)alloc((size_t)N2 * HN * 4);
    unsigned* m2    = (unsigned*)alloc((size_t)N2 * HN * 4);
    float*    den2  = (float*)alloc((size_t)N2 * HN * 4);
    float*    ex2   = (float*)alloc((size_t)E2 * HN * 4);
    float*    agg2  = (float*)alloc((size_t)N2 * HC * 4);
    float*    h2    = (float*)alloc((size_t)N2 * HC * 4);

    float*    xl3   = (float*)alloc((size_t)N2 * NOUT3 * 4);
    float*    skip3 = (float*)alloc((size_t)N3 * C_OUT * 4);
    float*    a_s3  = (float*)alloc((size_t)N2 * HN * 4);
    float*    a_d3  = (float*)alloc((size_t)N3 * HN * 4);
    unsigned* m3    = (unsigned*)alloc((size_t)N3 * HN * 4);
    float*    den3  = (float*)alloc((size_t)N3 * HN * 4);
    float*    ex3   = (float*)alloc((size_t)E3 * HN * 4);
    float*    agg3  = (float*)alloc((size_t)N3 * NOUT3 * 4);

    float* out = (float*)d_out;

    // ---- Layer 1 ----------------------------------------------------------
    launch_gemm(x, W1,  nullptr, xl1,   N0, HC, F_IN, stream);
    launch_gemm(x, Ws1, bs1,     skip1, N1, HC, F_IN, stream);
    gat_attention(xl1, N0, N1, src1, dst1, E1, as1, ad1,
                  a_s1, a_d1, m1, den1, ex1, agg1, C_HID, stream);
    {
        const long long tot = (long long)N1 * HC;
        finalize_elu_kernel<<<(unsigned)CDIV(tot, (long long)TPB), TPB, 0, stream>>>(
            agg1, b1, skip1, h1, tot, HC);
    }

    // ---- Layer 2 ----------------------------------------------------------
    launch_gemm(h1, W2,  nullptr, xl2,   N1, HC, HC, stream);
    launch_gemm(h1, Ws2, bs2,     skip2, N2, HC, HC, stream);
    gat_attention(xl2, N1, N2, src2, dst2, E2, as2, ad2,
                  a_s2, a_d2, m2, den2, ex2, agg2, C_HID, stream);
    {
        const long long tot = (long long)N2 * HC;
        finalize_elu_kernel<<<(unsigned)CDIV(tot, (long long)TPB), TPB, 0, stream>>>(
            agg2, b2, skip2, h2, tot, HC);
    }

    // ---- Layer 3 (mean over heads + log_softmax) --------------------------
    launch_gemm(h2, W3,  nullptr, xl3,   N2, NOUT3, HC, stream);
    launch_gemm(h2, Ws3, bs3,     skip3, N3, C_OUT, HC, stream);
    gat_attention(xl3, N2, N3, src3, dst3, E3, as3, ad3,
                  a_s3, a_d3, m3, den3, ex3, agg3, C_OUT, stream);
    finalize3_kernel<<<CDIV(N3, TPB), TPB, 0, stream>>>(agg3, b3, skip3, out,
                                                        N3, HN, C_OUT);
}